// TopK_16655883174209
// MI455X (gfx1250) — compile-verified
//
#include <hip/hip_runtime.h>
#include <stdint.h>

// Top-K=32 per row of 24576 fp32, ReLU, scatter into zeros.
// One workgroup per row; row staged in LDS via async global->LDS loads with
// staged s_wait_asynccnt draining (pass-0 histogram overlapped with loads),
// exact 4-pass radix select on order-preserving u32 keys, streaming NT output.

#define ROWLEN 24576
#define BLOCK  256
#define TOPK   32
#define NCOPY  32          // per-lane histogram copies (kills bank conflicts)
#define HPAD   257         // 257-dword stride => bank = (copy + bin) % 64
#define CHUNKS (ROWLEN / (BLOCK * 4))   // 24 x b128 per thread
#define MAXTIE 1024

typedef float v4f __attribute__((ext_vector_type(4)));

// monotonic float -> u32 key: larger float <=> larger unsigned key
__device__ __forceinline__ unsigned fkey(float f) {
    unsigned u = __float_as_uint(f);
    return u ^ (unsigned)(((int)u >> 31) | 0x80000000);
}

__global__ __launch_bounds__(BLOCK) void topk_scatter_kernel(
    const float* __restrict__ x, float* __restrict__ out) {

    __shared__ float    row[ROWLEN];            // 96 KB: whole row resident
    __shared__ unsigned hist[NCOPY * HPAD];     // 32.1 KB padded histograms
    __shared__ unsigned suffix[BLOCK];          // suffix sums over 256 bins
    __shared__ unsigned tieCols[MAXTIE];
    __shared__ unsigned tieSel[MAXTIE];
    __shared__ unsigned sh_digit, sh_krem, tieCount;

    const int tid = threadIdx.x;
    const int64_t rowBase = (int64_t)blockIdx.x * ROWLEN;
    const float* __restrict__ gin  = x   + rowBase;
    float*       __restrict__ gout = out + rowBase;

    // ---------- zero pass-0 histograms BEFORE issuing loads ----------
    for (int i = tid; i < NCOPY * HPAD; i += BLOCK) hist[i] = 0u;

    // ---------- issue all async global->LDS loads (single HBM read) ----------
    #pragma unroll
    for (int j = 0; j < CHUNKS; ++j) {
        int e = 4 * (tid + j * BLOCK);
        uint32_t laddr = (uint32_t)(uintptr_t)(&row[e]);   // low 32b = LDS offset
        uint64_t gaddr = (uint64_t)(uintptr_t)(gin + e);
        asm volatile("global_load_async_to_lds_b128 %0, %1, off"
                     :: "v"(laddr), "v"(gaddr) : "memory");
    }
    __syncthreads();   // hist zeroing visible; async loads still in flight

    unsigned prefix = 0;     // resolved high bits of the K-th largest key
    unsigned krem   = TOPK;  // how many still needed within current prefix
    unsigned* myhist = &hist[(tid & 31) * HPAD];   // wave32: lane-private copy

    for (int pass = 0; pass < 4; ++pass) {
        const int shift = 24 - 8 * pass;

        if (pass == 0) {
            // ---- fused drain + pass-0 histogram: ASYNCcnt retires in issue
            // order per wave, so after s_wait_asynccnt (CHUNKS-1-j) this
            // thread's chunk j is resident. Histogram work hides load latency.
            #pragma unroll
            for (int j = 0; j < CHUNKS; ++j) {
                asm volatile("s_wait_asynccnt %0" :: "i"(CHUNKS - 1 - j) : "memory");
                int e = 4 * (tid + j * BLOCK);
                v4f v = *(const v4f*)(&row[e]);            // own chunk only
                #pragma unroll
                for (int c = 0; c < 4; ++c)
                    atomicAdd(&myhist[fkey(v[c]) >> 24], 1u);
            }
        } else {
            const unsigned pmask = 0xFFFFFFFFu << (shift + 8);
            for (int i = tid; i < NCOPY * HPAD; i += BLOCK) hist[i] = 0u;
            __syncthreads();
            for (int j = 0; j < ROWLEN / BLOCK; ++j) {
                unsigned kk = fkey(row[tid + j * BLOCK]);
                if ((kk & pmask) == prefix)
                    atomicAdd(&myhist[(kk >> shift) & 0xFFu], 1u);
            }
        }
        __syncthreads();

        // reduce the 32 copies; lanes read distinct banks ((cp*257+tid)%64)
        unsigned c = 0;
        #pragma unroll
        for (int cp = 0; cp < NCOPY; ++cp) c += hist[cp * HPAD + tid];
        suffix[tid] = c;
        __syncthreads();

        // suffix[b] = #elements (matching prefix) with digit >= b
        for (int off = 1; off < BLOCK; off <<= 1) {
            unsigned v = (tid + off < BLOCK) ? suffix[tid + off] : 0u;
            __syncthreads();
            suffix[tid] += v;
            __syncthreads();
        }

        unsigned above = (tid == BLOCK - 1) ? 0u : suffix[tid + 1];
        if (suffix[tid] >= krem && above < krem) {   // exactly one thread fires
            sh_digit = (unsigned)tid;
            sh_krem  = krem - above;
        }
        __syncthreads();
        prefix |= sh_digit << shift;
        krem    = sh_krem;
        __syncthreads();
    }

    const unsigned T = prefix;    // exact key of the K-th largest value
    // krem >= 1 elements equal to T must be kept (ties broken by lowest column)

    // ---------- deterministic tie resolution ----------
    if (tid == 0) tieCount = 0u;
    __syncthreads();
    for (int j = 0; j < ROWLEN / BLOCK; ++j) {
        int col = tid + j * BLOCK;
        if (fkey(row[col]) == T) {
            unsigned p = atomicAdd(&tieCount, 1u);
            if (p < MAXTIE) tieCols[p] = (unsigned)col;
        }
    }
    __syncthreads();
    const unsigned nt = tieCount < MAXTIE ? tieCount : MAXTIE;
    for (unsigned i = tid; i < nt; i += BLOCK) {
        unsigned ci = tieCols[i], rank = 0;
        for (unsigned j2 = 0; j2 < nt; ++j2) rank += (tieCols[j2] < ci) ? 1u : 0u;
        tieSel[i] = (rank < krem) ? 1u : 0u;         // keep krem smallest columns
    }
    __syncthreads();

    // ---------- streaming output: zeros + ReLU'd selected values ----------
    for (int j = 0; j < CHUNKS; ++j) {
        int e = 4 * (tid + j * BLOCK);
        v4f v = *(const v4f*)(&row[e]);              // ds_load_b128
        v4f o;
        #pragma unroll
        for (int c = 0; c < 4; ++c) {
            float f = v[c];
            unsigned kk = fkey(f);
            bool inc = (kk > T);
            if (kk == T) {                           // rare path (ties)
                inc = false;
                unsigned col = (unsigned)(e + c);
                for (unsigned p = 0; p < nt; ++p)
                    if (tieCols[p] == col) { inc = (tieSel[p] != 0u); break; }
            }
            o[c] = inc ? fmaxf(f, 0.0f) : 0.0f;
        }
        __builtin_nontemporal_store(o, (v4f*)(gout + e));   // write-once: bypass L2
    }
}

extern "C" void kernel_launch(void* const* d_in, const int* in_sizes, int n_in,
                              void* d_out, int out_size, void* d_ws, size_t ws_size,
                              hipStream_t stream) {
    const float* x   = (const float*)d_in[0];
    float*       out = (float*)d_out;
    const int nrows  = in_sizes[0] / ROWLEN;   // 8192
    topk_scatter_kernel<<<dim3(nrows), dim3(BLOCK), 0, stream>>>(x, out);
}